// SMDSystemModel_39496519254133
// MI455X (gfx1250) — compile-verified
//
#include <hip/hip_runtime.h>
#include <math.h>

#ifndef __has_builtin
#define __has_builtin(x) 0
#endif

#if __has_builtin(__builtin_amdgcn_global_load_async_to_lds_b32) && \
    __has_builtin(__builtin_amdgcn_s_wait_asynccnt)
#define USE_ASYNC_LDS 1
#else
#define USE_ASYNC_LDS 0
#endif

#if __has_builtin(__builtin_amdgcn_rcpf)
#define FAST_RCP(x) __builtin_amdgcn_rcpf(x)
#else
#define FAST_RCP(x) (1.0f / (x))
#endif

#define BATCH  2048
#define BLK    32     // one wave32 per block
#define NBUF   8      // prefetch depth (6*8 = 48 outstanding async <= 63)
#define CCHUNK 32     // time chunks for the affine-map parallel scan

#if USE_ASYNC_LDS
// Builtin signature (probe-confirmed):
//   (addrspace(1) int* gaddr, addrspace(3) int* lds, imm offset, imm cpol)
typedef __attribute__((address_space(1))) int glb_int;
typedef __attribute__((address_space(3))) int lds_int;

// Issue 6 async global->LDS loads of row offb (bytes) into one lane's 32B slot.
__device__ __forceinline__ void stage6(const float* V, const float* I, const float* Ta,
                                       const float* F1, const float* F2, const float* F3,
                                       unsigned offb, float* row) {
    __builtin_amdgcn_global_load_async_to_lds_b32(
        (glb_int*)((const char*)V  + offb), (lds_int*)(row + 0), 0, 0);
    __builtin_amdgcn_global_load_async_to_lds_b32(
        (glb_int*)((const char*)I  + offb), (lds_int*)(row + 1), 0, 0);
    __builtin_amdgcn_global_load_async_to_lds_b32(
        (glb_int*)((const char*)Ta + offb), (lds_int*)(row + 2), 0, 0);
    __builtin_amdgcn_global_load_async_to_lds_b32(
        (glb_int*)((const char*)F1 + offb), (lds_int*)(row + 3), 0, 0);
    __builtin_amdgcn_global_load_async_to_lds_b32(
        (glb_int*)((const char*)F2 + offb), (lds_int*)(row + 4), 0, 0);
    __builtin_amdgcn_global_load_async_to_lds_b32(
        (glb_int*)((const char*)F3 + offb), (lds_int*)(row + 5), 0, 0);
}
#endif

__device__ __forceinline__ void read6(const float* row, float& Vv, float& Iv, float& Tav,
                                      float& F1v, float& F2v, float& F3v) {
    float4 a4 = *(const float4*)row;        // ds_load_b128
    float2 c2 = *(const float2*)(row + 4);  // ds_load_b64
    Vv = a4.x; Iv = a4.y; Tav = a4.z; F1v = a4.w; F2v = c2.x; F3v = c2.y;
}

__device__ __forceinline__ unsigned steps_from(const int* dt_p, const int* st_p, int T) {
    unsigned dtu = (unsigned)dt_p[0];
    unsigned s   = (unsigned)st_p[0] / dtu;
    if (s > (unsigned)T) s = (unsigned)T;
    return s;
}

struct SmdState { float T1, T2, T3, T4, T6, T8; };

// One explicit-Euler step + history stores (non-temporal streamed output).
__device__ __forceinline__ void smd_step(SmdState& s,
                                         float Vv, float Iv, float Tav,
                                         float F1v, float F2v, float F3v,
                                         float dtf,
                                         float* __restrict__ o0, float* __restrict__ o1,
                                         float* __restrict__ o2, float* __restrict__ o3,
                                         float* __restrict__ o4, float* __restrict__ o5,
                                         float* __restrict__ o6,
                                         unsigned offB) {
    const float c_p    = 4186.0f;
    const float c_p_sw = 4000.0f;
    const float inv60  = 1.0f / 60.0f;
    const float m4k    = 1025.0f / 60000.0f;
    const float invLeq = 1.0f / 100.0f;
    const float HoverL = 4.0f / 100.0f;
    const float invAsf = 1.0f / (500.0f * 1000.0f * 4186.0f);
    const float invRTV = 1.0f / (1000.0f * 10.0f);
    const float invcp  = 1.0f / 4186.0f;
    const float aA     = 10.0f;

    float m1   = F1v * inv60;
    float Tavg = 0.5f * (s.T1 + s.T2);
    float dT2  = (0.5f * Iv - HoverL * (Tavg - Tav)
                  - c_p * m1 * (s.T2 - s.T1) * invLeq) * invAsf;
    float T2n  = s.T2 + dT2 * dtf;

    float m1z = F1v * inv60;
    float m2  = (1.0f - Vv) * F2v * inv60;
    float m3  = F2v * inv60;
    float m4  = F3v * m4k;

    float dT3 = (m1z * T2n + m2 * s.T4 - (m1z + m2) * s.T3
                 - 100.0f * (s.T3 - Tav) * invcp) * invRTV;
    float dT4 = (m1z * s.T3 + m2 * s.T6 - (m1z + m2) * s.T4
                 - 100.0f * (s.T4 - Tav) * invcp) * invRTV;
    float T3n = s.T3 + dT3 * dtf;
    float T4n = s.T4 + dT4 * dtf;
    float T1n = T4n;

    float T5    = Vv * s.T6 + (1.0f - Vv) * T3n;
    float T7    = s.T8 - 4.0f;
    float rm3   = FAST_RCP(m3 * c_p);
    float rm4   = FAST_RCP(m4 * c_p_sw);
    float theta = aA * (rm3 - rm4);
    float e     = __expf(theta);
    float Ahe   = 1.0f - e;
    float eta2  = (m3 * c_p) * rm4;
    float Bhe   = eta2 * e;
    float eta1  = Ahe * FAST_RCP(1.1f - Bhe);
    float T6n   = T5 - 1.6f * eta1 * (T5 - T7);
    float T8n   = T7 + 0.45f * eta2 * (T5 - T6n);

    s.T1 = T1n; s.T2 = T2n; s.T3 = T3n; s.T4 = T4n; s.T6 = T6n; s.T8 = T8n;

    __builtin_nontemporal_store(T2n, (float*)((char*)o0 + offB));
    __builtin_nontemporal_store(T3n, (float*)((char*)o1 + offB));
    __builtin_nontemporal_store(T4n, (float*)((char*)o2 + offB));
    __builtin_nontemporal_store(T5,  (float*)((char*)o3 + offB));
    __builtin_nontemporal_store(T6n, (float*)((char*)o4 + offB));
    __builtin_nontemporal_store(T7,  (float*)((char*)o5 + offB));
    __builtin_nontemporal_store(T8n, (float*)((char*)o6 + offB));
}

// Compose one step's affine map into the running chunk map (rows over [x0;1]).
// Rows: R1,R2,R3,R4,R6,R8 each 7 coeffs. Verified against reference ordering:
// dT3 uses NEW T2n; dT4 uses OLD T3; T1n = T4n; T5 uses NEW T3n + OLD T6.
__device__ __forceinline__ void compose_step(
    float R1[7], float R2[7], float R3[7], float R4[7], float R6[7], float R8[7],
    float Vv, float Iv, float Tav, float F1v, float F2v, float F3v,
    float p02, float kq, float pc5, float pc04, float r, float ru) {
    float q   = F1v * kq;
    float a   = q - p02;
    float b   = 1.0f - p02 - q;
    float cc  = pc5 * Iv + pc04 * Tav;
    float m1z = F1v * (1.0f / 60.0f);
    float m2  = (1.0f - Vv) * F2v * (1.0f / 60.0f);
    float rm1 = r * m1z;
    float rm2 = r * m2;
    float g   = 1.0f - (rm1 + rm2) - ru;
    float c34 = ru * Tav;

    float m3c  = F2v * (4186.0f / 60.0f);
    float m4c  = F3v * (1025.0f * 4000.0f / 60000.0f);
    float rcm4 = FAST_RCP(m4c);
    float theta = 10.0f * (FAST_RCP(m3c) - rcm4);
    float e    = __expf(theta);
    float eta2 = m3c * rcm4;
    float eta1 = (1.0f - e) * FAST_RCP(1.1f - eta2 * e);
    float k6   = 1.6f * eta1;
    float w    = 0.45f * eta2 * k6;

    float nR2[7], nR3[7], nR4[7], R5[7], nR6[7], nR8[7];
#pragma unroll
    for (int j = 0; j < 7; ++j) nR2[j] = a * R1[j] + b * R2[j];
    nR2[6] += cc;
#pragma unroll
    for (int j = 0; j < 7; ++j) nR3[j] = rm1 * nR2[j] + rm2 * R4[j] + g * R3[j];
    nR3[6] += c34;
#pragma unroll
    for (int j = 0; j < 7; ++j) nR4[j] = rm1 * R3[j] + rm2 * R6[j] + g * R4[j];
    nR4[6] += c34;
#pragma unroll
    for (int j = 0; j < 7; ++j) R5[j] = nR3[j] + Vv * (R6[j] - nR3[j]);
#pragma unroll
    for (int j = 0; j < 7; ++j) {
        float r7 = R8[j] - ((j == 6) ? 4.0f : 0.0f);   // T7 = T8 - 4
        nR6[j] = R5[j] + k6 * (r7 - R5[j]);
        nR8[j] = r7 + w * (R5[j] - r7);
    }
#pragma unroll
    for (int j = 0; j < 7; ++j) {
        R1[j] = nR4[j]; R2[j] = nR2[j]; R3[j] = nR3[j];
        R4[j] = nR4[j]; R6[j] = nR6[j]; R8[j] = nR8[j];
    }
}

__device__ __forceinline__ void store_row(float* maps, int baseK, int B, int col,
                                          const float* R) {
#pragma unroll
    for (int j = 0; j < 7; ++j)
        __builtin_nontemporal_store(R[j], maps + (size_t)(baseK + j) * B + col);
}

// ---------------- Pass 1: per-chunk affine map composition ----------------
__global__ __launch_bounds__(BLK) void smd_pass1_maps(
    const float* __restrict__ V,  const float* __restrict__ I,
    const float* __restrict__ Ta, const float* __restrict__ F1,
    const float* __restrict__ F2, const float* __restrict__ F3,
    const int* __restrict__ dt_p, const int* __restrict__ st_p,
    float* __restrict__ maps, int T, int B, int L) {
    const int tid = threadIdx.x;
    const int col = blockIdx.x * BLK + tid;
    const int c   = blockIdx.y;

    const int   steps = (int)steps_from(dt_p, st_p, T);
    const float dtf   = (float)(unsigned)dt_p[0];

    int t0 = c * L;
    int t1 = t0 + L;
    if (t1 > steps) t1 = steps;
    const int n = (t1 > t0) ? (t1 - t0) : 0;

    // uniform per-step constants with dtf folded in
    const float p    = dtf * (1.0f / (500.0f * 1000.0f * 4186.0f));
    const float p02  = 0.02f * p;
    const float kq   = p * (4186.0f / 6000.0f);   // c_p/(60*100)
    const float pc5  = 0.5f * p;
    const float pc04 = 0.04f * p;
    const float r    = dtf * 1e-4f;               // dtf/(rho*TV)
    const float ru   = r * (100.0f / 4186.0f);

    // identity map
    float R1[7] = {}, R2[7] = {}, R3[7] = {}, R4[7] = {}, R6[7] = {}, R8[7] = {};
    R1[0] = 1.0f; R2[1] = 1.0f; R3[2] = 1.0f; R4[3] = 1.0f; R6[4] = 1.0f; R8[5] = 1.0f;

    const unsigned rowB = 4u * (unsigned)B;
    unsigned offBp = (unsigned)(t0 * B + col) * 4u;

#if USE_ASYNC_LDS
    __shared__ __align__(16) float buf[NBUF][BLK][8];
    const int n_pre = n < NBUF ? n : NBUF;
    for (int tp = 0; tp < n_pre; ++tp) { stage6(V, I, Ta, F1, F2, F3, offBp, &buf[tp][tid][0]); offBp += rowB; }

    int k = 0;
    float Vv, Iv, Tav, F1v, F2v, F3v;
    for (; k < n - NBUF; ++k) {
        __builtin_amdgcn_s_wait_asynccnt(6 * (NBUF - 1));
        const int sl = k & (NBUF - 1);
        read6(&buf[sl][tid][0], Vv, Iv, Tav, F1v, F2v, F3v);
        compose_step(R1, R2, R3, R4, R6, R8, Vv, Iv, Tav, F1v, F2v, F3v,
                     p02, kq, pc5, pc04, r, ru);
        stage6(V, I, Ta, F1, F2, F3, offBp, &buf[sl][tid][0]);
        offBp += rowB;
    }
    __builtin_amdgcn_s_wait_asynccnt(0);
    for (; k < n; ++k) {
        const int sl = k & (NBUF - 1);
        read6(&buf[sl][tid][0], Vv, Iv, Tav, F1v, F2v, F3v);
        compose_step(R1, R2, R3, R4, R6, R8, Vv, Iv, Tav, F1v, F2v, F3v,
                     p02, kq, pc5, pc04, r, ru);
    }
#else
    for (int k = 0; k < n; ++k) {
        float Vv  = __builtin_nontemporal_load((const float*)((const char*)V  + offBp));
        float Iv  = __builtin_nontemporal_load((const float*)((const char*)I  + offBp));
        float Tav = __builtin_nontemporal_load((const float*)((const char*)Ta + offBp));
        float F1v = __builtin_nontemporal_load((const float*)((const char*)F1 + offBp));
        float F2v = __builtin_nontemporal_load((const float*)((const char*)F2 + offBp));
        float F3v = __builtin_nontemporal_load((const float*)((const char*)F3 + offBp));
        compose_step(R1, R2, R3, R4, R6, R8, Vv, Iv, Tav, F1v, F2v, F3v,
                     p02, kq, pc5, pc04, r, ru);
        offBp += rowB;
    }
#endif
    // maps layout [C][42][B] (coalesced across lanes)
    const int base = c * 42;
    store_row(maps, base + 0 * 7, B, col, R1);
    store_row(maps, base + 1 * 7, B, col, R2);
    store_row(maps, base + 2 * 7, B, col, R3);
    store_row(maps, base + 3 * 7, B, col, R4);
    store_row(maps, base + 4 * 7, B, col, R6);
    store_row(maps, base + 5 * 7, B, col, R8);
}

// ------- Pass 2: fold chunk maps sequentially; emit chunk-start states -------
__global__ void smd_pass2_scan(const float* __restrict__ maps,
                               float* __restrict__ states, int B, int C) {
    const int col = blockIdx.x * blockDim.x + threadIdx.x;
    float x[6] = {50.0f, 60.0f, 60.0f, 50.0f, 25.0f, 60.0f};
    for (int c = 0; c < C; ++c) {
#pragma unroll
        for (int i = 0; i < 6; ++i)
            states[(size_t)(c * 6 + i) * B + col] = x[i];
        float y[6];
#pragma unroll
        for (int i = 0; i < 6; ++i) {
            const float* m = maps + (size_t)(c * 42 + i * 7) * B + col;
            float acc = m[(size_t)6 * B];
#pragma unroll
            for (int j = 0; j < 6; ++j) acc += m[(size_t)j * B] * x[j];
            y[i] = acc;
        }
#pragma unroll
        for (int i = 0; i < 6; ++i) x[i] = y[i];
    }
}

// --------- Pass 3: replay each chunk from its true start state ---------
__global__ __launch_bounds__(BLK) void smd_pass3_replay(
    const float* __restrict__ V,  const float* __restrict__ I,
    const float* __restrict__ Ta, const float* __restrict__ F1,
    const float* __restrict__ F2, const float* __restrict__ F3,
    const int* __restrict__ dt_p, const int* __restrict__ st_p,
    const float* __restrict__ states,
    float* __restrict__ out, int T, int B, int L) {
    const int tid = threadIdx.x;
    const int col = blockIdx.x * BLK + tid;
    const int c   = blockIdx.y;

    const int   steps = (int)steps_from(dt_p, st_p, T);
    const float dtf   = (float)(unsigned)dt_p[0];

    int t0 = c * L;
    int t1 = t0 + L;
    if (t1 > steps) t1 = steps;
    const int n = t1 - t0;
    if (n <= 0) return;

    SmdState s;
    s.T1 = states[(size_t)(c * 6 + 0) * B + col];
    s.T2 = states[(size_t)(c * 6 + 1) * B + col];
    s.T3 = states[(size_t)(c * 6 + 2) * B + col];
    s.T4 = states[(size_t)(c * 6 + 3) * B + col];
    s.T6 = states[(size_t)(c * 6 + 4) * B + col];
    s.T8 = states[(size_t)(c * 6 + 5) * B + col];

    const size_t TB = (size_t)T * (size_t)B;
    float* o0 = out + 0 * TB;  float* o1 = out + 1 * TB;
    float* o2 = out + 2 * TB;  float* o3 = out + 3 * TB;
    float* o4 = out + 4 * TB;  float* o5 = out + 5 * TB;
    float* o6 = out + 6 * TB;

    const unsigned rowB = 4u * (unsigned)B;
    unsigned offB  = (unsigned)(t0 * B + col) * 4u;
    unsigned offBp = offB;

#if USE_ASYNC_LDS
    __shared__ __align__(16) float buf[NBUF][BLK][8];
    const int n_pre = n < NBUF ? n : NBUF;
    for (int tp = 0; tp < n_pre; ++tp) { stage6(V, I, Ta, F1, F2, F3, offBp, &buf[tp][tid][0]); offBp += rowB; }

    int k = 0;
    float Vv, Iv, Tav, F1v, F2v, F3v;
    for (; k < n - NBUF; ++k) {
        __builtin_amdgcn_s_wait_asynccnt(6 * (NBUF - 1));
        const int sl = k & (NBUF - 1);
        read6(&buf[sl][tid][0], Vv, Iv, Tav, F1v, F2v, F3v);
        smd_step(s, Vv, Iv, Tav, F1v, F2v, F3v, dtf, o0, o1, o2, o3, o4, o5, o6, offB);
        stage6(V, I, Ta, F1, F2, F3, offBp, &buf[sl][tid][0]);
        offBp += rowB;
        offB  += rowB;
    }
    __builtin_amdgcn_s_wait_asynccnt(0);
    for (; k < n; ++k) {
        const int sl = k & (NBUF - 1);
        read6(&buf[sl][tid][0], Vv, Iv, Tav, F1v, F2v, F3v);
        smd_step(s, Vv, Iv, Tav, F1v, F2v, F3v, dtf, o0, o1, o2, o3, o4, o5, o6, offB);
        offB += rowB;
    }
#else
    for (int k = 0; k < n; ++k) {
        float Vv  = __builtin_nontemporal_load((const float*)((const char*)V  + offB));
        float Iv  = __builtin_nontemporal_load((const float*)((const char*)I  + offB));
        float Tav = __builtin_nontemporal_load((const float*)((const char*)Ta + offB));
        float F1v = __builtin_nontemporal_load((const float*)((const char*)F1 + offB));
        float F2v = __builtin_nontemporal_load((const float*)((const char*)F2 + offB));
        float F3v = __builtin_nontemporal_load((const float*)((const char*)F3 + offB));
        smd_step(s, Vv, Iv, Tav, F1v, F2v, F3v, dtf, o0, o1, o2, o3, o4, o5, o6, offB);
        offB += rowB;
    }
#endif
}

// --------- Fallback: original fused sequential kernel (64 waves) ---------
__global__ __launch_bounds__(BLK) void smd_scan_fused(
    const float* __restrict__ V,  const float* __restrict__ I,
    const float* __restrict__ Ta, const float* __restrict__ F1,
    const float* __restrict__ F2, const float* __restrict__ F3,
    const int* __restrict__ dt_p, const int* __restrict__ st_p,
    float* __restrict__ out, int T, int B) {
    const int tid = threadIdx.x;
    const int b   = blockIdx.x * BLK + tid;

    const int   steps = (int)steps_from(dt_p, st_p, T);
    const float dtf   = (float)(unsigned)dt_p[0];

    const size_t TB = (size_t)T * (size_t)B;
    float* o0 = out + 0 * TB;  float* o1 = out + 1 * TB;
    float* o2 = out + 2 * TB;  float* o3 = out + 3 * TB;
    float* o4 = out + 4 * TB;  float* o5 = out + 5 * TB;
    float* o6 = out + 6 * TB;

    const unsigned rowB = 4u * (unsigned)B;
    unsigned offB  = 4u * (unsigned)b;
    unsigned offBp = offB;

    SmdState s = {50.0f, 60.0f, 60.0f, 50.0f, 25.0f, 60.0f};

#if USE_ASYNC_LDS
    __shared__ __align__(16) float buf[NBUF][BLK][8];
    const int n_pre = steps < NBUF ? steps : NBUF;
    for (int tp = 0; tp < n_pre; ++tp) { stage6(V, I, Ta, F1, F2, F3, offBp, &buf[tp][tid][0]); offBp += rowB; }

    int t = 0;
    float Vv, Iv, Tav, F1v, F2v, F3v;
    for (; t < steps - NBUF; ++t) {
        __builtin_amdgcn_s_wait_asynccnt(6 * (NBUF - 1));
        const int sl = t & (NBUF - 1);
        read6(&buf[sl][tid][0], Vv, Iv, Tav, F1v, F2v, F3v);
        smd_step(s, Vv, Iv, Tav, F1v, F2v, F3v, dtf, o0, o1, o2, o3, o4, o5, o6, offB);
        stage6(V, I, Ta, F1, F2, F3, offBp, &buf[sl][tid][0]);
        offBp += rowB;
        offB  += rowB;
    }
    __builtin_amdgcn_s_wait_asynccnt(0);
    for (; t < steps; ++t) {
        const int sl = t & (NBUF - 1);
        read6(&buf[sl][tid][0], Vv, Iv, Tav, F1v, F2v, F3v);
        smd_step(s, Vv, Iv, Tav, F1v, F2v, F3v, dtf, o0, o1, o2, o3, o4, o5, o6, offB);
        offB += rowB;
    }
#else
    for (int t = 0; t < steps; ++t) {
        float Vv  = __builtin_nontemporal_load((const float*)((const char*)V  + offB));
        float Iv  = __builtin_nontemporal_load((const float*)((const char*)I  + offB));
        float Tav = __builtin_nontemporal_load((const float*)((const char*)Ta + offB));
        float F1v = __builtin_nontemporal_load((const float*)((const char*)F1 + offB));
        float F2v = __builtin_nontemporal_load((const float*)((const char*)F2 + offB));
        float F3v = __builtin_nontemporal_load((const float*)((const char*)F3 + offB));
        smd_step(s, Vv, Iv, Tav, F1v, F2v, F3v, dtf, o0, o1, o2, o3, o4, o5, o6, offB);
        offB += rowB;
    }
#endif
}

extern "C" void kernel_launch(void* const* d_in, const int* in_sizes, int n_in,
                              void* d_out, int out_size, void* d_ws, size_t ws_size,
                              hipStream_t stream) {
    const float* V  = (const float*)d_in[0];
    const float* I  = (const float*)d_in[1];
    const float* Ta = (const float*)d_in[2];
    const float* F1 = (const float*)d_in[3];
    const float* F2 = (const float*)d_in[4];
    const float* F3 = (const float*)d_in[5];
    const int*   dt = (const int*)d_in[6];
    const int*   st = (const int*)d_in[7];

    const int B = BATCH;
    const int T = in_sizes[0] / B;   // [T, B] time-major
    const int C = CCHUNK;
    const int L = (T + C - 1) / C;

    const size_t need = (size_t)C * 48 * B * sizeof(float); // 42 map + 6 state floats
    if (ws_size >= need) {
        float* maps   = (float*)d_ws;
        float* states = maps + (size_t)C * 42 * B;
        dim3 grid(B / BLK, C), block(BLK);
        smd_pass1_maps<<<grid, block, 0, stream>>>(V, I, Ta, F1, F2, F3, dt, st,
                                                   maps, T, B, L);
        smd_pass2_scan<<<dim3(B / 256), dim3(256), 0, stream>>>(maps, states, B, C);
        smd_pass3_replay<<<grid, block, 0, stream>>>(V, I, Ta, F1, F2, F3, dt, st,
                                                     states, (float*)d_out, T, B, L);
    } else {
        dim3 grid(B / BLK), block(BLK);
        smd_scan_fused<<<grid, block, 0, stream>>>(V, I, Ta, F1, F2, F3, dt, st,
                                                   (float*)d_out, T, B);
    }
}